// MorphologicalEquivariance_74285754352261
// MI455X (gfx1250) — compile-verified
//
#include <hip/hip_runtime.h>

#define EMB 128
#define LOG2_BOOST 0.5849625007211562f /* log2(1.5) */
#define EPS 1e-8f

typedef __attribute__((ext_vector_type(2))) float v2f;
typedef __attribute__((ext_vector_type(8))) float v8f;

// ---------------------------------------------------------------- zero scratch
__global__ void me_zero_kernel(int* __restrict__ counts, float* __restrict__ partials,
                               int R, int P) {
    int i = blockIdx.x * blockDim.x + threadIdx.x;
    if (i < R) counts[i] = 0;
    if (i < P) partials[i] = 0.0f;
}

// -------------------------------------------------- roots gather + root counts
__global__ void me_count_kernel(const int* __restrict__ wi, const int* __restrict__ rmap,
                                int* __restrict__ roots, int* __restrict__ counts, int n) {
    int i = blockIdx.x * blockDim.x + threadIdx.x;
    if (i < n) {
        int r = rmap[wi[i]];
        roots[i] = r;
        atomicAdd(&counts[r], 1);
    }
}

// ---------------------------------------------- per-root matvec via f32 WMMA
// block = 256 threads (8 waves), one block per root r.
// Wave w computes rows [16w, 16w+16) of f_r = M_r @ e_r + b_r using
// V_WMMA_F32_16X16X4_F32 with e_r packed into column 0 of the B matrix.
__global__ void __launch_bounds__(256)
me_root_feat_kernel(const float* __restrict__ emb, const float* __restrict__ mats,
                    const float* __restrict__ bias, float* __restrict__ f) {
    int r = blockIdx.x;
    __shared__ float se[EMB];
    if (threadIdx.x < EMB) se[threadIdx.x] = emb[r * EMB + threadIdx.x];
    __syncthreads();

    int wave = threadIdx.x >> 5;
    int lane = threadIdx.x & 31;
    int m0   = wave * 16;

    // A-matrix 16x4 f32 layout: lanes 0-15 -> K=0,1 ; lanes 16-31 -> K=2,3
    int  koff = (lane >= 16) ? 2 : 0;
    const float* Mrow = mats + (size_t)r * EMB * EMB
                             + (size_t)(m0 + (lane & 15)) * EMB + koff;
    // B-matrix 4x16: only column N=0 is nonzero (holds e_r); lanes 0 and 16 carry it
    float bmask = ((lane & 15) == 0) ? 1.0f : 0.0f;

    v8f c = {};
    for (int k = 0; k < EMB; k += 4) {
        v2f a = *(const v2f*)(Mrow + k);          // 8B-aligned global_load_b64
        v2f b;
        b.x = se[k + koff]     * bmask;
        b.y = se[k + koff + 1] * bmask;
        c = __builtin_amdgcn_wmma_f32_16x16x4_f32(
                /*neg_a=*/false, a, /*neg_b=*/false, b,
                /*c_mod=*/(short)0, c, /*reuse_a=*/false, /*reuse_b=*/false);
    }

    // C/D layout: VGPR v, lane l -> row (l<16 ? v : v+8), col l&15.  Col 0 lives
    // in lanes 0 (rows m0..m0+7) and 16 (rows m0+8..m0+15).
    if ((lane & 15) == 0) {
        int base = m0 + ((lane == 16) ? 8 : 0);
        #pragma unroll
        for (int v = 0; v < 8; ++v)
            f[r * EMB + base + v] = c[v] + bias[r * EMB + base + v];
    }
}

// ------------------------------------- boosted weights + per-block partial sum
__global__ void __launch_bounds__(256)
me_boost_kernel(const float* __restrict__ w, const int* __restrict__ roots,
                const int* __restrict__ counts, float* __restrict__ boosted,
                float* __restrict__ partials, int n) {
    __shared__ float red[256];
    int i = blockIdx.x * blockDim.x + threadIdx.x;
    float b = 0.0f;
    if (i < n) {
        float cm1 = (float)(counts[roots[i]] - 1);
        b = w[i] * exp2f(cm1 * LOG2_BOOST) + EPS;   // softmax(log(x)) == x / sum(x)
        boosted[i] = b;
    }
    red[threadIdx.x] = b;
    __syncthreads();
    #pragma unroll
    for (int s = 128; s > 0; s >>= 1) {
        if (threadIdx.x < s) red[threadIdx.x] += red[threadIdx.x + s];
        __syncthreads();
    }
    if (threadIdx.x == 0) partials[blockIdx.x] = red[0];
}

// ------------------------------------------------ deterministic final reduction
__global__ void me_reduce_kernel(const float* __restrict__ partials,
                                 float* __restrict__ sum, int P) {
    if (threadIdx.x == 0) {
        float s = 0.0f;
        for (int i = 0; i < P; ++i) s += partials[i];
        sum[0] = s;
    }
}

// ------------------------------- out[n,:] = f[root_n,:] * boosted[n]/sum  (f4)
__global__ void __launch_bounds__(128)
me_out_kernel(const float* __restrict__ f, const int* __restrict__ roots,
              const float* __restrict__ boosted, const float* __restrict__ sum,
              float4* __restrict__ out, int n) {
    int word = blockIdx.x * 4 + (threadIdx.x >> 5);   // 4 words per block
    int lane = threadIdx.x & 31;                      // 32 float4 per word (128 f32)
    if (word < n) {
        float scale = boosted[word] / sum[0];
        const float4* fr = (const float4*)(f + (size_t)roots[word] * EMB);
        float4 v = fr[lane];
        v.x *= scale; v.y *= scale; v.z *= scale; v.w *= scale;
        out[(size_t)word * (EMB / 4) + lane] = v;
    }
}

static inline size_t me_align256(size_t x) { return (x + 255) & ~(size_t)255; }

extern "C" void kernel_launch(void* const* d_in, const int* in_sizes, int n_in,
                              void* d_out, int out_size, void* d_ws, size_t ws_size,
                              hipStream_t stream) {
    const float* emb  = (const float*)d_in[0];   // (R, 128)
    const float* mats = (const float*)d_in[1];   // (R, 128, 128)
    const float* bias = (const float*)d_in[2];   // (R, 128)
    const float* attw = (const float*)d_in[3];   // (N,)
    const int*   wi   = (const int*)d_in[4];     // (N,)
    const int*   rmap = (const int*)d_in[5];     // (VOCAB,)

    const int N = in_sizes[3];
    const int R = in_sizes[0] / EMB;
    const int P = (N + 255) / 256;               // boost partial-sum blocks

    char* ws = (char*)d_ws;
    size_t off = 0;
    int*   counts   = (int*)(ws + off);   off = me_align256(off + (size_t)R * 4);
    int*   roots    = (int*)(ws + off);   off = me_align256(off + (size_t)N * 4);
    float* boosted  = (float*)(ws + off); off = me_align256(off + (size_t)N * 4);
    float* partials = (float*)(ws + off); off = me_align256(off + (size_t)P * 4);
    float* sum      = (float*)(ws + off); off = me_align256(off + 4);
    float* f        = (float*)(ws + off); off = me_align256(off + (size_t)R * EMB * 4);
    (void)ws_size; (void)n_in; (void)out_size;

    int zmax = (R > P) ? R : P;
    me_zero_kernel<<<(zmax + 255) / 256, 256, 0, stream>>>(counts, partials, R, P);
    me_count_kernel<<<(N + 255) / 256, 256, 0, stream>>>(wi, rmap, roots, counts, N);
    me_root_feat_kernel<<<R, 256, 0, stream>>>(emb, mats, bias, f);
    me_boost_kernel<<<P, 256, 0, stream>>>(attw, roots, counts, boosted, partials, N);
    me_reduce_kernel<<<1, 32, 0, stream>>>(partials, sum, P);
    me_out_kernel<<<(N + 3) / 4, 128, 0, stream>>>(f, roots, boosted, sum,
                                                   (float4*)d_out, N);
}